// PermutationMatrixLearningModule_19825569038350
// MI455X (gfx1250) — compile-verified
//
#include <hip/hip_runtime.h>

#define DD   64
#define NP1  65
#define BIGF 1e9f

#ifndef __has_builtin
#define __has_builtin(x) 0
#endif

#if __has_builtin(__builtin_amdgcn_global_load_async_to_lds_b128)
#define HAS_ALD 1
#else
#define HAS_ALD 0
#endif
#if __has_builtin(__builtin_amdgcn_global_store_async_from_lds_b128)
#define HAS_AST 1
#else
#define HAS_AST 0
#endif

// 16-byte payload type for the b128 async ops, in the address spaces the
// builtin expects: AS1 (global, prints as "__device__") and AS3 (LDS).
typedef int v4i __attribute__((ext_vector_type(4)));
typedef __attribute__((address_space(1))) v4i g_v4i;
typedef __attribute__((address_space(3))) v4i l_v4i;

__device__ __forceinline__ void wait_async0() {
#if __has_builtin(__builtin_amdgcn_s_wait_asynccnt)
  __builtin_amdgcn_s_wait_asynccnt(0);
#else
  asm volatile("s_wait_asynccnt 0x0" ::: "memory");
#endif
}

__device__ __forceinline__ float gumbel_of(float x) {
  // -log(-log(clip(x, 1e-20, 1 - 1e-7)))
  float xc = fminf(fmaxf(x, 1e-20f), 1.0f - 1e-7f);
  return -__logf(-__logf(xc));
}

// One wave (32 lanes) per sample. Lane l owns columns j=1+l and j=33+l
// of the (n+1)-sized JV arrays; column 0 is the virtual start column.
__global__ __launch_bounds__(32) void perm_sample_lap_kernel(
    const float* __restrict__ gamma,
    const float* __restrict__ u,
    float* __restrict__ out) {
  __shared__ __align__(16) float sC[DD * DD];  // cost matrix, then reused as one-hot out tile
  __shared__ float sU[NP1];                    // row duals (1-based rows)
  __shared__ int   sP[NP1];                    // p[j]: 1-based row matched to column j
  __shared__ int   sWay[NP1];

  const int l = threadIdx.x;
  const int b = blockIdx.x;
  const float* ub = u + (size_t)b * (DD * DD);
  float* ob = out + (size_t)b * (DD * DD);

  float4* c4 = (float4*)sC;
  const float4* g4 = (const float4*)gamma;

  // ---- Stage u[b] (16 KB) into LDS via CDNA5 async global->LDS path ----
#if HAS_ALD
  #pragma unroll 4
  for (int k = 0; k < 32; ++k) {
    int c = l + 32 * k;  // float4 chunk index, 0..1023
    __builtin_amdgcn_global_load_async_to_lds_b128(
        (g_v4i*)(ub + 4 * c), (l_v4i*)(sC + 4 * c), 0, 0);
  }
  wait_async0();
#else
  const float4* u4 = (const float4*)ub;
  for (int k = 0; k < 32; ++k) {
    int c = l + 32 * k;
    c4[c] = u4[c];
  }
#endif
  __syncthreads();

  // ---- In-place transform: raw uniforms -> cost = -(gamma + gumbel) ----
  for (int k = 0; k < 32; ++k) {
    int c = l + 32 * k;
    float4 uu = c4[c];
    float4 gg = g4[c];
    float4 r;
    r.x = -(gg.x + gumbel_of(uu.x));
    r.y = -(gg.y + gumbel_of(uu.y));
    r.z = -(gg.z + gumbel_of(uu.z));
    r.w = -(gg.w + gumbel_of(uu.w));
    c4[c] = r;
  }
  // ---- Init duals / matching ----
  for (int idx = l; idx < NP1; idx += 32) {
    sU[idx] = 0.0f;
    sP[idx] = 0;
    sWay[idx] = 0;
  }
  float v1 = 0.0f, v2 = 0.0f;  // column duals for j=1+l, j=33+l
  __syncthreads();

  // ---- Jonker-Volgenant: insert rows 1..64 ----
  for (int i = 0; i < DD; ++i) {
    if (l == 0) sP[0] = i + 1;
    float minv1 = BIGF, minv2 = BIGF;
    int way1 = 0, way2 = 0;
    bool used1 = false, used2 = false;
    const int pc1 = sP[1 + l];   // p is constant during the search phase
    const int pc2 = sP[33 + l];
    int j0 = 0;

    while (true) {
      // used[j0] = true
      if (j0 == 1 + l) used1 = true;
      else if (j0 == 33 + l) used2 = true;
      // (j0 == 0 handled implicitly: lane 0 always bumps sU[i+1] below)

      const int i0 = sP[j0];        // broadcast LDS read (uniform address)
      const float ui0 = sU[i0];
      const int row = i0 - 1;
      const float cA = sC[row * DD + l]      - ui0 - v1;  // column j=1+l
      const float cB = sC[row * DD + 32 + l] - ui0 - v2;  // column j=33+l
      if (!used1 && cA < minv1) { minv1 = cA; way1 = j0; }
      if (!used2 && cB < minv2) { minv2 = cB; way2 = j0; }

      // masked argmin over columns 1..64 (first-min tie break, like jnp.argmin)
      float m1 = used1 ? BIGF : minv1;
      float m2 = used2 ? BIGF : minv2;
      float bv; int bj;
      if (m1 <= m2) { bv = m1; bj = 1 + l; } else { bv = m2; bj = 33 + l; }
      #pragma unroll
      for (int off = 16; off > 0; off >>= 1) {
        float ov = __shfl_xor(bv, off, 32);
        int   oj = __shfl_xor(bj, off, 32);
        if (ov < bv || (ov == bv && oj < bj)) { bv = ov; bj = oj; }
      }
      const float delta = bv;   // uniform
      const int j1 = bj;        // uniform

      // dual updates: rows of used columns are pairwise distinct -> plain LDS RMW safe
      if (l == 0) sU[i + 1] += delta;                      // virtual column 0, p[0]=i+1
      if (used1) { sU[pc1] += delta; v1 -= delta; } else { minv1 -= delta; }
      if (used2) { sU[pc2] += delta; v2 -= delta; } else { minv2 -= delta; }

      j0 = j1;
      if (sP[j0] == 0) break;   // reached an unmatched column
    }

    // ---- augment along 'way' chain (lane 0, sequential pointer chase) ----
    sWay[1 + l] = way1;
    sWay[33 + l] = way2;
    __syncthreads();
    if (l == 0) {
      int j = j0;
      while (j != 0) {
        int jn = sWay[j];
        sP[j] = sP[jn];
        j = jn;
      }
    }
    __syncthreads();
  }

  // ---- Build one-hot tile in LDS (reuse sC), then drain to global ----
  float4 z4 = make_float4(0.0f, 0.0f, 0.0f, 0.0f);
  for (int k = 0; k < 32; ++k) c4[l + 32 * k] = z4;
  __syncthreads();
  {
    int r1 = sP[1 + l] - 1;   // 0-based row matched to column l
    int r2 = sP[33 + l] - 1;  // 0-based row matched to column 32+l
    sC[r1 * DD + l] = 1.0f;
    sC[r2 * DD + 32 + l] = 1.0f;
  }
  __syncthreads();

#if HAS_AST
  #pragma unroll 4
  for (int k = 0; k < 32; ++k) {
    int c = l + 32 * k;
    __builtin_amdgcn_global_store_async_from_lds_b128(
        (g_v4i*)(ob + 4 * c), (l_v4i*)(sC + 4 * c), 0, 0);
  }
  wait_async0();
#else
  float4* o4 = (float4*)ob;
  for (int k = 0; k < 32; ++k) {
    int c = l + 32 * k;
    o4[c] = c4[c];
  }
#endif
}

extern "C" void kernel_launch(void* const* d_in, const int* in_sizes, int n_in,
                              void* d_out, int out_size, void* d_ws, size_t ws_size,
                              hipStream_t stream) {
  (void)n_in; (void)out_size; (void)d_ws; (void)ws_size;
  const float* gamma = (const float*)d_in[0];
  const float* u     = (const float*)d_in[1];
  float* out = (float*)d_out;
  const int nsamples = in_sizes[1] / (DD * DD);  // 8192
  if (nsamples <= 0) return;
  perm_sample_lap_kernel<<<nsamples, 32, 0, stream>>>(gamma, u, out);
}